// Pool3d_54640573939779
// MI455X (gfx1250) — compile-verified
//
#include <hip/hip_runtime.h>
#include <hip/hip_bf16.h>
#include <stdint.h>

// Segmented scatter-max pooling (decimation max-pool).
//   out[seg, c] = max over rows r with vt_map[r]==seg of in[r, c]; empty -> 0
//
// Memory-bound (0.2 flop/byte): WMMA is inapplicable; the MI455X-relevant
// paths are async global->LDS staging (ASYNCcnt) and native L2 atomics.
// Float max is done via the monotone uint encoding + global_atomic_max_u32
// (single native VMEM atomic on gfx1250; 64MB output is L2-resident in 192MB L2).

#define CCH 64            // channels
#define TILE_ROWS 64      // rows staged per block (64 * 256B = 16KB LDS)
#define BLOCK 256
#define CHUNKS_PER_ROW 16 // 16 x float4 = 64 floats
#define CHUNKS_PER_TILE (TILE_ROWS * CHUNKS_PER_ROW)   // 1024
#define CHUNKS_PER_THREAD (CHUNKS_PER_TILE / BLOCK)    // 4

#define ENC_NEG_INF 0x007FFFFFu   // enc(-inf) = ~0xFF800000

#if __has_builtin(__builtin_amdgcn_global_load_async_to_lds_b128)
#define HAVE_ASYNC_LDS 1
#else
#define HAVE_ASYNC_LDS 0
#endif

// 4 x i32 vector type matching the builtin's expected pointee
// (diagnostic showed: "__attribute__((__vector_size__(4 * sizeof(int)))) int __device__ *")
typedef int vi4 __attribute__((vector_size(16)));
typedef vi4 __attribute__((address_space(1)))* vi4_gptr;
typedef vi4 __attribute__((address_space(3)))* vi4_lptr;

__device__ __forceinline__ unsigned enc_f32(float f) {
  unsigned u = __float_as_uint(f);
  return (u & 0x80000000u) ? ~u : (u | 0x80000000u);
}
__device__ __forceinline__ float dec_f32(unsigned u) {
  unsigned v = (u & 0x80000000u) ? (u & 0x7fffffffu) : ~u;
  return __uint_as_float(v);
}

__device__ __forceinline__ void wait_async_zero() {
#if __has_builtin(__builtin_amdgcn_s_wait_asynccnt)
  __builtin_amdgcn_s_wait_asynccnt(0);
#else
  asm volatile("s_wait_asynccnt 0" ::: "memory");
#endif
}

// ---------------- Pass 1: init output to encoded -inf ----------------
__global__ void pool_init_kernel(uint4* __restrict__ out, int n4) {
  int i = blockIdx.x * blockDim.x + threadIdx.x;
  if (i < n4) out[i] = make_uint4(ENC_NEG_INF, ENC_NEG_INF, ENC_NEG_INF, ENC_NEG_INF);
}

// ---------------- Pass 2: scatter-max via native u32 atomics ----------------
__global__ void pool_scatter_kernel(const float* __restrict__ in,
                                    const long long* __restrict__ vt_map,
                                    unsigned* __restrict__ out) {
  __shared__ float     lds_data[TILE_ROWS * CCH]; // 16 KB
  __shared__ long long lds_seg[TILE_ROWS];

  const int tid = threadIdx.x;
  const long long row_base = (long long)blockIdx.x * TILE_ROWS;
  const float* gbase = in + row_base * CCH;

#if HAVE_ASYNC_LDS
  // Stage the whole tile LDS-side with the CDNA5 async data path.
  #pragma unroll
  for (int i = 0; i < CHUNKS_PER_THREAD; ++i) {
    int chunk = tid + i * BLOCK;
    vi4* gsrc = (vi4*)(gbase + (size_t)chunk * 4);   // generic, const dropped
    vi4* ldst = (vi4*)(&lds_data[chunk * 4]);        // generic
    __builtin_amdgcn_global_load_async_to_lds_b128(
        (vi4_gptr)gsrc, (vi4_lptr)ldst, /*imm offset*/ 0, /*cpol*/ 0);
  }
#endif
  // Overlap: stage segment ids while the async DMA runs.
  if (tid < TILE_ROWS) lds_seg[tid] = vt_map[row_base + tid];
#if HAVE_ASYNC_LDS
  wait_async_zero();
#endif
  __syncthreads();

  #pragma unroll
  for (int i = 0; i < CHUNKS_PER_THREAD; ++i) {
    int chunk = tid + i * BLOCK;
    int row = chunk >> 4;            // chunk / 16
    int c4  = (chunk & 15) * 4;      // starting channel of this float4
    long long seg = lds_seg[row];
    float4 v;
#if HAVE_ASYNC_LDS
    v = *(const float4*)(&lds_data[row * CCH + c4]);
#else
    v = *(const float4*)(gbase + (size_t)chunk * 4);
#endif
    if (seg >= 0) {
      unsigned* o = out + (size_t)seg * CCH + c4;
      atomicMax(o + 0, enc_f32(v.x));
      atomicMax(o + 1, enc_f32(v.y));
      atomicMax(o + 2, enc_f32(v.z));
      atomicMax(o + 3, enc_f32(v.w));
    }
  }
}

// ---------------- Pass 3: decode; empty segments (-inf) -> 0 ----------------
__global__ void pool_finalize_kernel(unsigned* __restrict__ buf, int n4) {
  int i = blockIdx.x * blockDim.x + threadIdx.x;
  if (i < n4) {
    uint4 u = ((const uint4*)buf)[i];
    float4 f;
    f.x = (u.x == ENC_NEG_INF) ? 0.0f : dec_f32(u.x);
    f.y = (u.y == ENC_NEG_INF) ? 0.0f : dec_f32(u.y);
    f.z = (u.z == ENC_NEG_INF) ? 0.0f : dec_f32(u.z);
    f.w = (u.w == ENC_NEG_INF) ? 0.0f : dec_f32(u.w);
    ((float4*)buf)[i] = f;
  }
}

extern "C" void kernel_launch(void* const* d_in, const int* in_sizes, int n_in,
                              void* d_out, int out_size, void* d_ws, size_t ws_size,
                              hipStream_t stream) {
  const float*     in       = (const float*)d_in[0];
  // d_in[1] = vt_replace (int64) -- unused for max pooling
  const long long* vt_map   = (const long long*)d_in[2];
  // d_in[3] = vt_out scalar -- implied by out_size
  unsigned* out_u = (unsigned*)d_out;

  const int n_rows = in_sizes[2];            // N_IN (vt_map element count)
  const int n4     = out_size / 4;           // output as uint4/float4 chunks

  const int init_blocks = (n4 + BLOCK - 1) / BLOCK;
  pool_init_kernel<<<init_blocks, BLOCK, 0, stream>>>((uint4*)out_u, n4);

  const int scatter_blocks = n_rows / TILE_ROWS;  // 1048576 / 64 = 16384
  pool_scatter_kernel<<<scatter_blocks, BLOCK, 0, stream>>>(in, vt_map, out_u);

  pool_finalize_kernel<<<init_blocks, BLOCK, 0, stream>>>(out_u, n4);
}